// GCN_46411416601218
// MI455X (gfx1250) — compile-verified
//
#include <hip/hip_runtime.h>
#include <hip/hip_bf16.h>

// ---------------------------------------------------------------------------
// GCN forward for MI455X (gfx1250, wave32).
// Roofline: edge scatter dominates (~6.8 GB L2 traffic, ~290us @ 23.3 TB/s);
// GEMMs (16.5 GFLOP) run on V_WMMA_F32_16X16X4_F32 (exact fp32). This round:
// fuse bias+self-loop into the GEMM epilogue and ReLU into the consumers,
// removing ~0.7 GB of elementwise traffic, with zero extra workspace.
// ---------------------------------------------------------------------------

typedef float v2f __attribute__((ext_vector_type(2)));
typedef float v8f __attribute__((ext_vector_type(8)));

static constexpr int NN = 50000;
static constexpr int NE = 800000;
static constexpr int G  = 100;
static constexpr int C1 = 512;
static constexpr int C2 = 256;
static constexpr int C3 = 128;

// ---------------------------- degree / norm --------------------------------

__global__ void k_deg_init(float* deg) {
    int i = blockIdx.x * blockDim.x + threadIdx.x;
    if (i < NN) deg[i] = 1.0f;                    // self-loop contribution
}

__global__ void k_deg_accum(const int* __restrict__ dst, float* __restrict__ deg) {
    int e = blockIdx.x * blockDim.x + threadIdx.x;
    if (e < NE) atomicAdd(&deg[dst[e]], 1.0f);
}

__global__ void k_dis(float* deg) {               // deg -> deg^{-1/2} in place
    int i = blockIdx.x * blockDim.x + threadIdx.x;
    if (i < NN) deg[i] = rsqrtf(deg[i]);          // deg >= 1 always (self loop)
}

// ------------- layer-1 projection (K=3) + fused bias/self-loop -------------
// m[n,c]   = sum_k x[n,k] W[k,c]
// agg[n,c] = b[c] + dis[n]^2 * m[n,c]     (scatter then atomically adds edges)

__global__ void k_gemm_k3(const float* __restrict__ x, const float* __restrict__ W,
                          const float* __restrict__ b, const float* __restrict__ dis,
                          float* __restrict__ m, float* __restrict__ agg) {
    int t = blockIdx.x * blockDim.x + threadIdx.x;          // one float4 of C1
    if (t >= NN * (C1 / 4)) return;
    int n  = t / (C1 / 4);
    int c4 = (t % (C1 / 4)) * 4;
    float x0 = x[n * 3 + 0], x1 = x[n * 3 + 1], x2 = x[n * 3 + 2];
    float d  = dis[n];
    float d2 = d * d;
    float4 w0 = *(const float4*)(W + 0 * C1 + c4);
    float4 w1 = *(const float4*)(W + 1 * C1 + c4);
    float4 w2 = *(const float4*)(W + 2 * C1 + c4);
    float4 bb = *(const float4*)(b + c4);
    float4 r, a;
    r.x = x0 * w0.x + x1 * w1.x + x2 * w2.x;
    r.y = x0 * w0.y + x1 * w1.y + x2 * w2.y;
    r.z = x0 * w0.z + x1 * w1.z + x2 * w2.z;
    r.w = x0 * w0.w + x1 * w1.w + x2 * w2.w;
    a.x = bb.x + d2 * r.x;  a.y = bb.y + d2 * r.y;
    a.z = bb.z + d2 * r.z;  a.w = bb.w + d2 * r.w;
    *(float4*)(m   + (size_t)n * C1 + c4) = r;
    *(float4*)(agg + (size_t)n * C1 + c4) = a;
}

// ------------------------ fp32 WMMA GEMM (M x K) @ (K x N) -----------------
// Block: 256 threads = 8 waves. Block tile: 16 (M) x 128 (N); each wave owns
// one 16x16 tile. K stepped in chunks of 32 (8 wmma per chunk).
// reluA: apply fmaxf(.,0) on A-tile load (folds previous layer's ReLU).
// Fused epilogue: m = A@B and agg = bias + dis^2 * m, both written here.
// Requires M%16==0, N%128==0, K%32==0 (true for all call sites).

__global__ void k_gemm_wmma(const float* __restrict__ A, const float* __restrict__ B,
                            const float* __restrict__ bias, const float* __restrict__ dis,
                            float* __restrict__ Cm, float* __restrict__ agg,
                            int M, int N, int K, int reluA) {
    __shared__ float As[16][36];     // 16 x 32 (+pad, keeps float2 reads 8B-aligned)
    __shared__ float Bs[32][132];    // 32 x 128 (+pad vs 64-bank conflicts)

    const int tid  = threadIdx.x;
    const int wave = tid >> 5;
    const int lane = tid & 31;
    const int m0     = blockIdx.x * 16;
    const int nBlock = blockIdx.y * 128;
    const int n0     = nBlock + wave * 16;

    const int rowA = lane & 15;      // A frag: lanes 0-15 hold K=0,1; 16-31 hold K=2,3
    const int kh   = lane >> 4;
    const int colB = lane & 15;      // B frag mirrors A's K-half split

    v8f acc = {};

    for (int k0 = 0; k0 < K; k0 += 32) {
        // Cooperative tile loads (coalesced).
#pragma unroll
        for (int r = 0; r < 2; ++r) {            // A tile: 512 floats
            int idx = tid + r * 256;
            float v = A[(size_t)(m0 + (idx >> 5)) * K + k0 + (idx & 31)];
            if (reluA) v = fmaxf(v, 0.0f);
            As[idx >> 5][idx & 31] = v;
        }
#pragma unroll
        for (int r = 0; r < 16; ++r) {           // B tile: 4096 floats
            int idx = tid + r * 256;
            Bs[idx >> 7][idx & 127] = B[(size_t)(k0 + (idx >> 7)) * N + nBlock + (idx & 127)];
        }
        __syncthreads();

#pragma unroll
        for (int kk = 0; kk < 32; kk += 4) {
            v2f a, b;
            a.x = As[rowA][kk + 2 * kh];
            a.y = As[rowA][kk + 2 * kh + 1];
            b.x = Bs[kk + 2 * kh][wave * 16 + colB];
            b.y = Bs[kk + 2 * kh + 1][wave * 16 + colB];
            acc = __builtin_amdgcn_wmma_f32_16x16x4_f32(
                false, a, false, b, (short)0, acc, false, false);
        }
        __syncthreads();
    }

    // C/D layout: VGPR r -> row r (lanes 0-15) / row r+8 (lanes 16-31), col = lane&15.
    const int col = n0 + (lane & 15);
    const float bcol = bias[col];
#pragma unroll
    for (int r = 0; r < 8; ++r) {
        int row = m0 + r + 8 * (lane >> 4);
        float d  = dis[row];
        float mv = acc[r];
        Cm [(size_t)row * N + col] = mv;
        agg[(size_t)row * N + col] = bcol + d * d * mv;
    }
}

// ------------------------- edge scatter ------------------------------------
// One wave per edge: agg[dst] += dis[src]*dis[dst] * m[src]  (f32 L2 atomics;
// float4 gathers: 512B/wave per iteration).

__global__ void k_scatter(const float* __restrict__ m, const float* __restrict__ dis,
                          const int* __restrict__ src, const int* __restrict__ dst,
                          float* __restrict__ agg, int Cc) {
    int wid  = (blockIdx.x * blockDim.x + threadIdx.x) >> 5;
    int lane = threadIdx.x & 31;
    if (wid >= NE) return;
    int s = src[wid], d = dst[wid];
    float nrm = dis[s] * dis[d];
    const float4* mrow = (const float4*)(m + (size_t)s * Cc);
    float* arow = agg + (size_t)d * Cc;
    for (int c4 = lane; c4 < (Cc >> 2); c4 += 32) {
        float4 v = mrow[c4];
        atomicAdd(&arow[c4 * 4 + 0], nrm * v.x);
        atomicAdd(&arow[c4 * 4 + 1], nrm * v.y);
        atomicAdd(&arow[c4 * 4 + 2], nrm * v.z);
        atomicAdd(&arow[c4 * 4 + 3], nrm * v.w);
    }
}

// ------------------------------- pooling + head ----------------------------

__global__ void k_pool_zero(float* p) {           // pooled[G*C3] ++ counts[G], contiguous
    int t = blockIdx.x * blockDim.x + threadIdx.x;
    if (t < G * C3 + G) p[t] = 0.0f;
}

// ReLU folded into the pooling read (h3 = relu(agg3)).
__global__ void k_pool(const float* __restrict__ h, const int* __restrict__ batch,
                       float* __restrict__ pooled, float* __restrict__ counts) {
    int t = blockIdx.x * blockDim.x + threadIdx.x;
    int n = t >> 5, lane = t & 31;
    if (n >= NN) return;
    int g = batch[n];
    for (int c = lane; c < C3; c += 32)
        atomicAdd(&pooled[g * C3 + c], fmaxf(h[(size_t)n * C3 + c], 0.0f));
    if (lane == 0) atomicAdd(&counts[g], 1.0f);
}

__global__ void k_head(const float* __restrict__ pooled, const float* __restrict__ counts,
                       const float* __restrict__ Wc, const float* __restrict__ bc,
                       float* __restrict__ out) {
    int g = blockIdx.x * blockDim.x + threadIdx.x;
    if (g >= G) return;
    float inv = 1.0f / fmaxf(counts[g], 1.0f);
    float l0 = bc[0], l1 = bc[1];
    for (int c = 0; c < C3; ++c) {
        float p = pooled[g * C3 + c] * inv;
        l0 += p * Wc[c * 2 + 0];
        l1 += p * Wc[c * 2 + 1];
    }
    float mx  = fmaxf(l0, l1);
    float lse = mx + logf(expf(l0 - mx) + expf(l1 - mx));
    out[g * 2 + 0] = l0 - lse;
    out[g * 2 + 1] = l1 - lse;
}

// ------------------------------- launcher ----------------------------------

extern "C" void kernel_launch(void* const* d_in, const int* in_sizes, int n_in,
                              void* d_out, int out_size, void* d_ws, size_t ws_size,
                              hipStream_t stream) {
    const float* x  = (const float*)d_in[0];
    const float* W1 = (const float*)d_in[1];
    const float* b1 = (const float*)d_in[2];
    const float* W2 = (const float*)d_in[3];
    const float* b2 = (const float*)d_in[4];
    const float* W3 = (const float*)d_in[5];
    const float* b3 = (const float*)d_in[6];
    const float* Wc = (const float*)d_in[7];
    const float* bc = (const float*)d_in[8];
    const int*   ei = (const int*)d_in[9];      // [2, NE] row-major: src then dst
    const int* batch = (const int*)d_in[10];
    float* out = (float*)d_out;

    const int* esrc = ei;
    const int* edst = ei + NE;

    // Workspace layout (205 MB total, same as non-fused plan):
    //   big0 [0, 102.4MB)     big1 [102.4MB, 204.8MB)   dis/pooled/counts after.
    // L1: m1->big0, agg1->big1        | scatter big0 -> big1
    // L2: A=big1, m2->big0lo, agg2->big0hi | scatter big0lo -> big0hi
    // L3: A=big0hi, m3->big1lo, agg3->big1mid | scatter big1lo -> big1mid
    char* ws = (char*)d_ws;
    const size_t bigBytes = (size_t)NN * C1 * sizeof(float);   // 102.4 MB
    float* big0 = (float*)(ws);
    float* big1 = (float*)(ws + bigBytes);
    float* dis    = (float*)(ws + 2 * bigBytes);               // NN floats
    float* pooled = dis + NN;                                  // G*C3
    float* counts = pooled + G * C3;                           // G

    float* m1   = big0;
    float* agg1 = big1;                         // = h1 after scatter (+relu on load)
    float* m2   = big0;                         // 50000*256 = 51.2 MB (low half)
    float* agg2 = big0 + (size_t)NN * C2;       // high half of big0
    float* m3   = big1;                         // 50000*128 = 25.6 MB
    float* agg3 = big1 + (size_t)NN * C3;

    const int T = 256;
    auto blocks = [](long long n, int t) { return (unsigned)((n + t - 1) / t); };

    // --- degree / normalization ---
    k_deg_init <<<blocks(NN, T), T, 0, stream>>>(dis);
    k_deg_accum<<<blocks(NE, T), T, 0, stream>>>(edst, dis);
    k_dis      <<<blocks(NN, T), T, 0, stream>>>(dis);

    // --- layer 1: project (K=3) + fused bias/self-loop, scatter ---
    k_gemm_k3  <<<blocks((long long)NN * (C1 / 4), T), T, 0, stream>>>(x, W1, b1, dis, m1, agg1);
    k_scatter  <<<blocks((long long)NE * 32, T), T, 0, stream>>>(m1, dis, esrc, edst, agg1, C1);

    // --- layer 2: WMMA GEMM 50000x512 @ 512x256 (relu on load, fused epilogue) ---
    k_gemm_wmma<<<dim3(NN / 16, C2 / 128), T, 0, stream>>>(agg1, W2, b2, dis, m2, agg2,
                                                           NN, C2, C1, 1);
    k_scatter  <<<blocks((long long)NE * 32, T), T, 0, stream>>>(m2, dis, esrc, edst, agg2, C2);

    // --- layer 3: WMMA GEMM 50000x256 @ 256x128 ---
    k_gemm_wmma<<<dim3(NN / 16, C3 / 128), T, 0, stream>>>(agg2, W3, b3, dis, m3, agg3,
                                                           NN, C3, C2, 1);
    k_scatter  <<<blocks((long long)NE * 32, T), T, 0, stream>>>(m3, dis, esrc, edst, agg3, C3);

    // --- mean pool (relu on read) + classifier head + log_softmax ---
    k_pool_zero<<<blocks(G * C3 + G, T), T, 0, stream>>>(pooled);
    k_pool     <<<blocks((long long)NN * 32, T), T, 0, stream>>>(agg3, batch, pooled, counts);
    k_head     <<<1, 128, 0, stream>>>(pooled, counts, Wc, bc, out);
}